// GCNEncoder_65644280152109
// MI455X (gfx1250) — compile-verified
//
#include <hip/hip_runtime.h>

#define GCN_IN  256
#define GCN_HID 64

typedef __attribute__((ext_vector_type(2))) float v2f;
typedef __attribute__((ext_vector_type(8))) float v8f;

// ---------------- degree / normalization ----------------

__global__ __launch_bounds__(256) void k_zero_i32(int* __restrict__ p, int n) {
  int i = blockIdx.x * blockDim.x + threadIdx.x;
  if (i < n) p[i] = 0;
}

__global__ __launch_bounds__(256) void k_deg_count(const int* __restrict__ col,
                                                   int* __restrict__ deg, int E) {
  int e = blockIdx.x * blockDim.x + threadIdx.x;
  if (e < E) atomicAdd(&deg[col[e]], 1);
}

// deg (int) -> dinv (float) in place; +1 folds in the self loop, so deg>0 always.
__global__ __launch_bounds__(256) void k_dinv(int* __restrict__ degf, int n) {
  int i = blockIdx.x * blockDim.x + threadIdx.x;
  if (i < n) {
    float d = (float)degf[i] + 1.0f;
    degf[i] = __float_as_int(rsqrtf(d));
  }
}

// ---------------- GEMM: xt = x @ W via V_WMMA_F32_16X16X4_F32 ----------------
// One wave per 16-row tile of x; 4 accumulators cover all 64 output columns.
// A 16x4 f32 layout: VGPR0 = K {0|2} (lane<16 | lane>=16), VGPR1 = K {1|3}.
// B 4x16 f32 layout mirrored; C/D: VGPR r = rows {r | 8+r}, col = lane&15.

__global__ __launch_bounds__(128) void k_gemm_wmma(const float* __restrict__ x,
                                                   const float* __restrict__ Wm,
                                                   float* __restrict__ xt, int N) {
  const int lane = threadIdx.x & 31;
  const int mt   = blockIdx.x * (blockDim.x >> 5) + (threadIdx.x >> 5);
  const int m0   = mt * 16;
  if (m0 >= N) return;

  if (m0 + 16 <= N) {                       // full tile (wave-uniform branch)
    const int mrow  = m0 + (lane & 15);
    const int khalf = (lane >> 4) << 1;     // 0 or 2
    const int nc    = lane & 15;
    v8f acc0 = {}, acc1 = {}, acc2 = {}, acc3 = {};

    for (int k = 0; k < GCN_IN; k += 4) {
      v2f a = *(const v2f*)(x + (size_t)mrow * GCN_IN + (k + khalf));
      const float* wp = Wm + (size_t)(k + khalf) * GCN_HID + nc;
      v2f b0, b1, b2, b3;
      b0.x = wp[0];  b0.y = wp[GCN_HID + 0];
      b1.x = wp[16]; b1.y = wp[GCN_HID + 16];
      b2.x = wp[32]; b2.y = wp[GCN_HID + 32];
      b3.x = wp[48]; b3.y = wp[GCN_HID + 48];
      acc0 = __builtin_amdgcn_wmma_f32_16x16x4_f32(false, a, false, b0, (short)0, acc0, false, false);
      acc1 = __builtin_amdgcn_wmma_f32_16x16x4_f32(false, a, false, b1, (short)0, acc1, false, false);
      acc2 = __builtin_amdgcn_wmma_f32_16x16x4_f32(false, a, false, b2, (short)0, acc2, false, false);
      acc3 = __builtin_amdgcn_wmma_f32_16x16x4_f32(false, a, false, b3, (short)0, acc3, false, false);
    }

    const int mrowD = m0 + ((lane >> 4) << 3);   // rows r (lanes 0-15) / 8+r (lanes 16-31)
#pragma unroll
    for (int r = 0; r < 8; ++r) {
      float* o = xt + (size_t)(mrowD + r) * GCN_HID + nc;
      o[0]  = acc0[r];
      o[16] = acc1[r];
      o[32] = acc2[r];
      o[48] = acc3[r];
    }
  } else {                                   // scalar tail (unused when N%16==0)
    for (int m = m0 + lane; m < N; m += 32) {
      for (int n = 0; n < GCN_HID; ++n) {
        float s = 0.f;
        for (int k = 0; k < GCN_IN; ++k)
          s += x[(size_t)m * GCN_IN + k] * Wm[(size_t)k * GCN_HID + n];
        xt[(size_t)m * GCN_HID + n] = s;
      }
    }
  }
}

// ---------------- self-loop init: out = xt * dinv^2 ----------------

__global__ __launch_bounds__(256) void k_self_init(const float* __restrict__ xt,
                                                   const float* __restrict__ dinv,
                                                   float* __restrict__ out, int total) {
  int idx = blockIdx.x * blockDim.x + threadIdx.x;
  if (idx < total) {
    float di = dinv[idx >> 6];               // idx / 64
    out[idx] = xt[idx] * di * di;
  }
}

// ---------------- edge scatter: one wave per edge, 2 channels per lane ----------------

__global__ __launch_bounds__(256) void k_scatter(const int* __restrict__ row,
                                                 const int* __restrict__ col,
                                                 const float* __restrict__ xt,
                                                 const float* __restrict__ dinv,
                                                 float* __restrict__ out, int E) {
  int gid  = blockIdx.x * blockDim.x + threadIdx.x;
  int e    = gid >> 5;                        // wave-uniform
  if (e >= E) return;
  int lane = gid & 31;
  int r = __builtin_amdgcn_readfirstlane(row[e]);
  int c = __builtin_amdgcn_readfirstlane(col[e]);
  float nrm = dinv[r] * dinv[c];
  v2f m = *(const v2f*)(xt + (size_t)r * GCN_HID + lane * 2);   // coalesced b64 gather
  float* o = out + (size_t)c * GCN_HID + lane * 2;
  unsafeAtomicAdd(o,     m.x * nrm);          // hardware global_atomic_add_f32
  unsafeAtomicAdd(o + 1, m.y * nrm);
}

// ---------------- bias + ReLU ----------------

__global__ __launch_bounds__(256) void k_finalize(float* __restrict__ out,
                                                  const float* __restrict__ b, int total) {
  int idx = blockIdx.x * blockDim.x + threadIdx.x;
  if (idx < total) {
    float v = out[idx] + b[idx & (GCN_HID - 1)];
    out[idx] = v > 0.f ? v : 0.f;
  }
}

// ---------------- launcher ----------------

extern "C" void kernel_launch(void* const* d_in, const int* in_sizes, int n_in,
                              void* d_out, int out_size, void* d_ws, size_t ws_size,
                              hipStream_t stream) {
  const float* x  = (const float*)d_in[0];
  const int*   ei = (const int*)d_in[1];
  const float* W  = (const float*)d_in[2];
  const float* b  = (const float*)d_in[3];
  float* out = (float*)d_out;

  const int HID = in_sizes[3];              // 64
  const int IN  = in_sizes[2] / HID;        // 256
  const int N   = in_sizes[0] / IN;         // 100000
  const int E   = in_sizes[1] / 2;          // 3200000
  const int* row = ei;                      // edge_index[0]
  const int* col = ei + E;                  // edge_index[1]

  // workspace layout: [ xt : N*HID floats ][ deg/dinv : N ints/floats ]
  float* xt  = (float*)d_ws;
  int*   deg = (int*)((char*)d_ws + (size_t)N * HID * sizeof(float));
  float* dinv = (float*)deg;                // written in place by k_dinv

  const int T = 256;
  k_zero_i32 <<<(N + T - 1) / T, T, 0, stream>>>(deg, N);
  k_deg_count<<<(E + T - 1) / T, T, 0, stream>>>(col, deg, E);
  k_dinv     <<<(N + T - 1) / T, T, 0, stream>>>(deg, N);

  const int mtiles = (N + 15) / 16;         // one wave per tile, 4 waves/block
  k_gemm_wmma<<<(mtiles + 3) / 4, 128, 0, stream>>>(x, W, xt, N);

  const int total = N * HID;
  k_self_init<<<(total + T - 1) / T, T, 0, stream>>>(xt, dinv, out, total);

  const long long sthreads = (long long)E * 32;
  k_scatter  <<<(int)((sthreads + T - 1) / T), T, 0, stream>>>(row, col, xt, dinv, out, E);

  k_finalize <<<(total + T - 1) / T, T, 0, stream>>>(out, b, total);
}